// Net_57939108823457
// MI455X (gfx1250) — compile-verified
//
#include <hip/hip_runtime.h>
#include <hip/hip_bf16.h>
#include <math.h>

typedef float v2f __attribute__((ext_vector_type(2)));
typedef float v8f __attribute__((ext_vector_type(8)));

#define DFEAT 64

// ---------------- utility kernels ----------------

__global__ void fill_kernel(float* __restrict__ p, float v, size_t n) {
    size_t i = (size_t)blockIdx.x * blockDim.x + threadIdx.x;
    if (i < n) p[i] = v;
}

// deg[dst[e]] += 1  (deg pre-filled with 1.0 for the self loop)
__global__ void deg_kernel(const int* __restrict__ dst, float* __restrict__ deg, int E) {
    int e = blockIdx.x * blockDim.x + threadIdx.x;
    if (e < E) atomicAdd(&deg[dst[e]], 1.0f);
}

__global__ void rsqrt_kernel(float* __restrict__ p, int n) {
    int i = blockIdx.x * blockDim.x + threadIdx.x;
    if (i < n) p[i] = rsqrtf(p[i]);   // deg >= 1 always (self loops)
}

// out[i][j] = b[j]  (bias pre-seed before the scatter-add)
__global__ void bias_init_kernel(float* __restrict__ out, const float* __restrict__ b, size_t total) {
    size_t i = (size_t)blockIdx.x * blockDim.x + threadIdx.x;
    if (i < total) out[i] = b[i & (DFEAT - 1)];
}

__global__ void relu_kernel(float* __restrict__ p, size_t n) {
    size_t i = (size_t)blockIdx.x * blockDim.x + threadIdx.x;
    if (i < n) p[i] = fmaxf(p[i], 0.0f);
}

// ---------------- dense GEMM via V_WMMA_F32_16X16X4_F32 ----------------
// C[M,64] = A[M,K] @ W[K,64].  grid.x = M/16 row stripes, 128 thr = 4 waves,
// wave w owns N-tile [16w, 16w+16).  K stepped by 4 per WMMA.
__global__ void gemm_wmma_f32(const float* __restrict__ A, const float* __restrict__ W,
                              float* __restrict__ C, int M, int K) {
    const int lane = threadIdx.x & 31;
    const int wv   = threadIdx.x >> 5;      // 0..3 -> N tile
    const int m0   = blockIdx.x * 16;
    const int n0   = wv * 16;
    const int l15  = lane & 15;
    const int hi   = lane >> 4;             // 0 or 1
    const int kb   = hi * 2;                // A: lanes16-31 hold K=2,3 of each step

    int mrow = m0 + l15; if (mrow >= M) mrow = M - 1;   // clamp (M%16==0 normally)
    const int ncol = n0 + l15;
    const float* __restrict__ arow = A + (size_t)mrow * K;

    v8f acc = {};
    for (int k = 0; k < K; k += 4) {
        v2f a, b;
        a.x = arow[k + kb];
        a.y = arow[k + kb + 1];
        b.x = W[(size_t)(k + kb)     * DFEAT + ncol];
        b.y = W[(size_t)(k + kb + 1) * DFEAT + ncol];
        acc = __builtin_amdgcn_wmma_f32_16x16x4_f32(
                  false, a, false, b, (short)0, acc, false, false);
    }
    // D layout: vgpr r -> row m0 + r + 8*hi, col = n0 + (lane&15)
    const int crow0 = m0 + hi * 8;
#pragma unroll
    for (int r = 0; r < 8; ++r) {
        int row = crow0 + r;
        if (row < M) C[(size_t)row * DFEAT + ncol] = acc[r];
    }
}

// ---------------- edge aggregation: out[dst] += h[src] * dinv[src]*dinv[dst] ----
// 16 threads per (edge or self-loop), float4 each -> full 256B row coalesced.
__global__ void aggregate_kernel(const float* __restrict__ h,
                                 const int* __restrict__ src,
                                 const int* __restrict__ dst,
                                 const float* __restrict__ dinv,
                                 float* __restrict__ out, int E, int N) {
    long long t = (long long)blockIdx.x * blockDim.x + threadIdx.x;
    int edge = (int)(t >> 4);
    int part = ((int)t & 15) * 4;
    if (edge >= E + N) return;
    int s, d;
    if (edge < E) { s = src[edge]; d = dst[edge]; }
    else          { s = d = edge - E; }          // self loop
    float w = dinv[s] * dinv[d];
    const float4 hv = *reinterpret_cast<const float4*>(h + (size_t)s * DFEAT + part);
    float* o = out + (size_t)d * DFEAT + part;
    atomicAdd(o + 0, hv.x * w);
    atomicAdd(o + 1, hv.y * w);
    atomicAdd(o + 2, hv.z * w);
    atomicAdd(o + 3, hv.w * w);
}

// ---------------- pooling: pool[g][0:64]=sum, pool[g][64:128]=max -------------
__device__ inline void atomicMaxF(float* addr, float v) {
    if (v >= 0.0f) atomicMax((int*)addr, __float_as_int(v));
    else           atomicMin((unsigned int*)addr, __float_as_uint(v));
}

__global__ void pool_init_kernel(float* __restrict__ pool, int G) {
    int i = blockIdx.x * blockDim.x + threadIdx.x;
    if (i < G * 2 * DFEAT) pool[i] = ((i & (2 * DFEAT - 1)) < DFEAT) ? 0.0f : -INFINITY;
}

__global__ void pool_kernel(const float* __restrict__ h, const int* __restrict__ batch,
                            float* __restrict__ pool, int N) {
    long long t = (long long)blockIdx.x * blockDim.x + threadIdx.x;
    int node = (int)(t >> 4);
    int f = ((int)t & 15) * 4;
    if (node >= N) return;
    int g = batch[node];
    const float4 hv = *reinterpret_cast<const float4*>(h + (size_t)node * DFEAT + f);
    float* sums = pool + (size_t)g * (2 * DFEAT) + f;
    float* maxs = sums + DFEAT;
    atomicAdd(sums + 0, hv.x);  atomicAdd(sums + 1, hv.y);
    atomicAdd(sums + 2, hv.z);  atomicAdd(sums + 3, hv.w);
    atomicMaxF(maxs + 0, hv.x); atomicMaxF(maxs + 1, hv.y);
    atomicMaxF(maxs + 2, hv.z); atomicMaxF(maxs + 3, hv.w);
}

// ---------------- tiny MLP head ----------------
__global__ void mlp_dense_kernel(const float* __restrict__ in, const float* __restrict__ W,
                                 const float* __restrict__ b, float* __restrict__ out,
                                 int K, int Nout, int relu) {
    int g = blockIdx.x, j = threadIdx.x;
    if (j >= Nout) return;
    float acc = b[j];
    for (int k = 0; k < K; ++k) acc += in[(size_t)g * K + k] * W[(size_t)k * Nout + j];
    if (relu) acc = fmaxf(acc, 0.0f);
    out[(size_t)g * Nout + j] = acc;
}

__global__ void mlp_out_kernel(const float* __restrict__ in, const float* __restrict__ w,
                               const float* __restrict__ b, float* __restrict__ out, int G) {
    int g = threadIdx.x;
    if (g >= G) return;
    float acc = b[0];
    for (int k = 0; k < DFEAT; ++k) acc += in[(size_t)g * DFEAT + k] * w[k];
    out[g] = acc;
}

// ---------------- launcher ----------------
extern "C" void kernel_launch(void* const* d_in, const int* in_sizes, int n_in,
                              void* d_out, int out_size, void* d_ws, size_t ws_size,
                              hipStream_t stream) {
    const float* x      = (const float*)d_in[0];
    const int*   eidx   = (const int*)  d_in[1];
    const int*   batch  = (const int*)  d_in[2];
    const float* W_in   = (const float*)d_in[3];
    const float* b_in   = (const float*)d_in[4];
    const float* W_hid  = (const float*)d_in[5];
    const float* b_hid  = (const float*)d_in[6];
    const float* lin1_w = (const float*)d_in[7];
    const float* lin1_b = (const float*)d_in[8];
    const float* lin2_w = (const float*)d_in[9];
    const float* lin2_b = (const float*)d_in[10];
    const float* lin3_w = (const float*)d_in[11];
    const float* lin3_b = (const float*)d_in[12];
    float* out = (float*)d_out;

    const int N   = in_sizes[2];
    const int E   = in_sizes[1] / 2;
    const int CIN = in_sizes[0] / N;
    const int G   = out_size;                 // output is [G,1]
    const int L   = in_sizes[6] / DFEAT + 1;  // b_hid holds (L-1)*D
    const int* src = eidx;
    const int* dst = eidx + E;

    // carve workspace (float-aligned, 256B-rounded slices)
    float* wsf = (float*)d_ws;
    size_t off = 0;
    auto carve = [&](size_t nel) { float* p = wsf + off; off += (nel + 63) & ~(size_t)63; return p; };
    float* dinv = carve((size_t)N);
    float* T    = carve((size_t)N * DFEAT);   // post-GEMM h
    float* S0   = carve((size_t)N * DFEAT);   // node state ping
    float* S1   = carve((size_t)N * DFEAT);   // node state pong
    float* pool = carve((size_t)G * 2 * DFEAT);
    float* m1   = carve((size_t)G * DFEAT);
    float* m2   = carve((size_t)G * DFEAT);
    (void)ws_size; (void)n_in;

    const int TB = 256;
    // degree -> dinv
    fill_kernel<<<(N + TB - 1) / TB, TB, 0, stream>>>(dinv, 1.0f, (size_t)N);
    deg_kernel<<<(E + TB - 1) / TB, TB, 0, stream>>>(dst, dinv, E);
    rsqrt_kernel<<<(N + TB - 1) / TB, TB, 0, stream>>>(dinv, N);

    const size_t nd = (size_t)N * DFEAT;
    const long long aggT = (long long)(E + N) * 16;
    const float* cur = x;
    int K = CIN;
    for (int l = 0; l < L; ++l) {
        const float* Wl = (l == 0) ? W_in : W_hid + (size_t)(l - 1) * DFEAT * DFEAT;
        const float* bl = (l == 0) ? b_in : b_hid + (size_t)(l - 1) * DFEAT;
        float* Snext = (l & 1) ? S1 : S0;

        gemm_wmma_f32<<<(N + 15) / 16, 128, 0, stream>>>(cur, Wl, T, N, K);
        bias_init_kernel<<<(int)((nd + TB - 1) / TB), TB, 0, stream>>>(Snext, bl, nd);
        aggregate_kernel<<<(int)((aggT + TB - 1) / TB), TB, 0, stream>>>(T, src, dst, dinv, Snext, E, N);
        if (l != L - 1)
            relu_kernel<<<(int)((nd + TB - 1) / TB), TB, 0, stream>>>(Snext, nd);
        cur = Snext;
        K = DFEAT;
    }

    // global add + max pooling
    pool_init_kernel<<<(G * 2 * DFEAT + TB - 1) / TB, TB, 0, stream>>>(pool, G);
    pool_kernel<<<(int)(((long long)N * 16 + TB - 1) / TB), TB, 0, stream>>>(cur, batch, pool, N);

    // MLP head
    mlp_dense_kernel<<<G, DFEAT, 0, stream>>>(pool, lin1_w, lin1_b, m1, 2 * DFEAT, DFEAT, 1);
    mlp_dense_kernel<<<G, DFEAT, 0, stream>>>(m1, lin2_w, lin2_b, m2, DFEAT, DFEAT, 1);
    mlp_out_kernel<<<1, DFEAT, 0, stream>>>(m2, lin3_w, lin3_b, out, G);
}